// PairwiseConv_22488448761917
// MI455X (gfx1250) — compile-verified
//
#include <hip/hip_runtime.h>

// ---------------------------------------------------------------------------
// PairwiseConv on MI455X (gfx1250, wave32, WMMA).
//
// Refactor: y[b,:,n] = ( cnt_i[n]*(W0 @ x[b,:,n] + bias) + W1 @ s[b,:,n] ) / max(deg_j[n],1)
//           s[b,:,n] = sum_{e: idx_i[e]==n} x[b,:,idx_j[e]]
//           out[b,127,n] = deg_j[n]
// Edge work becomes an L2-resident scatter-accumulate (8 MB targets vs 192 MB
// L2); the GEMM drops from 34 GFLOP (edge space) to 1 GFLOP (node space) and
// runs on the full-precision fp32 WMMA V_WMMA_F32_16X16X4_F32.
// v2: 8 waves per block share their mTile's A tiles (W0/W1 16x128) via LDS,
// cutting redundant A-side global loads 8x and feeding WMMA from ds_load_b64.
// ---------------------------------------------------------------------------

typedef float v2f __attribute__((ext_vector_type(2)));
typedef float v8f __attribute__((ext_vector_type(8)));

#define BATCH 4
#define CIN   128
#define NCOL  4096
#define NEDGE 131072
#define CCONV 127
#define COUT  128

// workspace layout (float offsets)
#define OFF_S     0                        // BATCH*CIN*NCOL accumulated neighbor sums
#define SZ_S      (BATCH*CIN*NCOL)
#define OFF_CNT   (OFF_S + SZ_S)           // NCOL : bincount(idx_i)
#define OFF_DEG   (OFF_CNT + NCOL)         // NCOL : bincount(idx_j)
#define OFF_W0    (OFF_DEG + NCOL)         // 128x128 dense W[:, :, 0], row 127 = 0
#define OFF_W1    (OFF_W0 + CIN*CIN)       // 128x128 dense W[:, :, 1], row 127 = 0
#define OFF_B     (OFF_W1 + CIN*CIN)       // 128 padded bias
#define WS_FLOATS (OFF_B + COUT)

__global__ void zero_ws(float* __restrict__ ws) {
  int i = blockIdx.x * blockDim.x + threadIdx.x;
  if (i < WS_FLOATS) ws[i] = 0.0f;
}

// Densify W (C_CONV,CIN,2) -> two row-major 128x128 matrices, pad row 127 with 0.
__global__ void prep_w(const float* __restrict__ W, const float* __restrict__ b,
                       float* __restrict__ ws) {
  int t = blockIdx.x * blockDim.x + threadIdx.x;   // 0 .. 128*128-1
  int o = t >> 7, c = t & 127;
  float w0 = 0.0f, w1 = 0.0f;
  if (o < CCONV) {
    w0 = W[(o * CIN + c) * 2 + 0];
    w1 = W[(o * CIN + c) * 2 + 1];
  }
  ws[OFF_W0 + o * CIN + c] = w0;
  ws[OFF_W1 + o * CIN + c] = w1;
  if (c == 0) ws[OFF_B + o] = (o < CCONV) ? b[o] : 0.0f;
}

// One thread per (edge, channel): s[b,c,idx_i] += x[b,c,idx_j]; counts once per edge.
// All targets are L2-resident (s = 8 MB), so the 67M f32 atomics stay in L2.
__global__ void edge_aggregate(const float* __restrict__ x,
                               const int* __restrict__ idx_i,
                               const int* __restrict__ idx_j,
                               float* __restrict__ ws) {
  int t = blockIdx.x * blockDim.x + threadIdx.x;   // NEDGE*CIN threads
  int e = t >> 7;
  int c = t & 127;
  int ni = idx_i[e];
  int nj = idx_j[e];
  if (c == 0) {
    atomicAdd(ws + OFF_CNT + ni, 1.0f);
    atomicAdd(ws + OFF_DEG + nj, 1.0f);
  }
  float* s = ws + OFF_S;
#pragma unroll
  for (int bb = 0; bb < BATCH; ++bb) {
    float v = x[(bb * CIN + c) * NCOL + nj];
    atomicAdd(s + (bb * CIN + c) * NCOL + ni, v);
  }
}

// 256-thread block = 8 waves = 8 consecutive nTiles of one (mTile, batch).
// The block's A tiles (W0/W1 rows mTile*16..+15, all K) are staged in LDS once
// and shared by all 8 waves; each wave runs K=128 in 32 steps of
// V_WMMA_F32_16X16X4_F32 with two independent accumulators (W0@x, W1@s).
__global__ __launch_bounds__(256) void gemm_combine(const float* __restrict__ x,
                                                    const float* __restrict__ ws_c,
                                                    float* __restrict__ out) {
  const float* s   = ws_c + OFF_S;
  const float* w0  = ws_c + OFF_W0;
  const float* w1  = ws_c + OFF_W1;
  const float* bp  = ws_c + OFF_B;
  const float* cnt = ws_c + OFF_CNT;
  const float* deg = ws_c + OFF_DEG;

  __shared__ float w0t[16 * CIN];   // A tile of W0: [row within mTile][K]
  __shared__ float w1t[16 * CIN];   // A tile of W1
  __shared__ float bt[16];          // bias slice

  const int tid   = threadIdx.x;
  const int mTile = blockIdx.y;     // 0..7
  const int bb    = blockIdx.z;     // 0..3

  // Cooperative stage: 2*16*128 floats across 256 threads (b64-friendly strides).
  for (int i = tid; i < 16 * CIN; i += 256) {
    const int r = i >> 7, c = i & 127;
    w0t[i] = w0[(mTile * 16 + r) * CIN + c];
    w1t[i] = w1[(mTile * 16 + r) * CIN + c];
  }
  if (tid < 16) bt[tid] = bp[mTile * 16 + tid];
  __syncthreads();

  const int lane  = tid & 31;       // full wave32, EXEC all ones (WMMA requirement)
  const int wave  = tid >> 5;       // 0..7 -> nTile within block
  const int m     = lane & 15;
  const int kh    = lane >> 4;      // lane-half selects K half (A/B) and M half (C/D)
  const int nTile = blockIdx.x * 8 + wave;
  const int n     = nTile * 16 + m;

  const float* xb = x + bb * CIN * NCOL;
  const float* sb = s + bb * CIN * NCOL;

  v8f acc0 = {};  // W0 @ x tile
  v8f acc1 = {};  // W1 @ s tile
#pragma unroll 4
  for (int k4 = 0; k4 < CIN / 4; ++k4) {
    const int kb = k4 * 4 + kh * 2;           // even -> 8B-aligned LDS reads
    // A 16x4 fragments from LDS (shared across the 8 waves)
    v2f a0 = *(const v2f*)&w0t[m * CIN + kb];
    v2f a1 = *(const v2f*)&w1t[m * CIN + kb];
    // B 4x16 fragments from global: col = lane&15; lanes>=16 carry K rows {2,3}
    v2f bx, bs;
    bx.x = xb[(kb + 0) * NCOL + n]; bx.y = xb[(kb + 1) * NCOL + n];
    bs.x = sb[(kb + 0) * NCOL + n]; bs.y = sb[(kb + 1) * NCOL + n];
    acc0 = __builtin_amdgcn_wmma_f32_16x16x4_f32(false, a0, false, bx,
                                                 (short)0, acc0, false, false);
    acc1 = __builtin_amdgcn_wmma_f32_16x16x4_f32(false, a1, false, bs,
                                                 (short)0, acc1, false, false);
  }

  const float cn    = cnt[n];
  const float dn    = deg[n];
  const float denom = (dn == 0.0f) ? 1.0f : dn;
#pragma unroll
  for (int r = 0; r < 8; ++r) {
    const int row = r + kh * 8;                      // C/D: VGPR r -> row r + 8*laneHalf
    const int o   = mTile * 16 + row;
    float v = (cn * (acc0[r] + bt[row]) + acc1[r]) / denom;
    if (o == CCONV) v = dn;                          // counts channel
    out[(bb * COUT + o) * NCOL + n] = v;
  }
}

extern "C" void kernel_launch(void* const* d_in, const int* in_sizes, int n_in,
                              void* d_out, int out_size, void* d_ws, size_t ws_size,
                              hipStream_t stream) {
  const float* x     = (const float*)d_in[0];
  const float* W     = (const float*)d_in[1];
  const float* b     = (const float*)d_in[2];
  const int*   idx_i = (const int*)d_in[3];
  const int*   idx_j = (const int*)d_in[4];
  float* ws  = (float*)d_ws;
  float* out = (float*)d_out;

  zero_ws<<<(WS_FLOATS + 255) / 256, 256, 0, stream>>>(ws);
  prep_w<<<(CIN * COUT + 255) / 256, 256, 0, stream>>>(W, b, ws);
  edge_aggregate<<<(NEDGE * CIN) / 256, 256, 0, stream>>>(x, idx_i, idx_j, ws);
  gemm_combine<<<dim3(NCOL / (16 * 8), COUT / 16, BATCH), 256, 0, stream>>>(x, ws, out);
}